// GraphEncoderHGT_67422396613283
// MI455X (gfx1250) — compile-verified
//
#include <hip/hip_runtime.h>
#include <math.h>

#define HID 128

typedef __attribute__((ext_vector_type(2))) float v2f;
typedef __attribute__((ext_vector_type(8))) float v8f;

// ---- monotonic float<->uint encoding for atomicMax-based segment max ----
__device__ __forceinline__ unsigned fenc(float f) {
  unsigned u = __float_as_uint(f);
  return (u & 0x80000000u) ? ~u : (u | 0x80000000u);
}
__device__ __forceinline__ float fdec(unsigned u) {
  return __uint_as_float((u & 0x80000000u) ? (u & 0x7fffffffu) : ~u);
}
// fenc(-inf) == 0x007FFFFF
#define ENC_NEG_INF 0x007FFFFFu

// =====================================================================
// GEMM: Y[N,128] = X[N,128] @ W[128,128] + bias  (mode 0)
// mode 1: Y = relu( s*(X@W+b) + (1-s)*Hskip ),  s = sigmoid(*skipGate)
// One wave computes a 16x128 slab via V_WMMA_F32_16X16X4_F32.
// W staged in LDS, packed so B-fragments are 8B ds loads:
//   Wlds[((k>>2)*128 + n)*4 + (k&3)] = W[k][n]
// =====================================================================
__global__ __launch_bounds__(256) void gemm128_kernel(
    const float* __restrict__ X, const float* __restrict__ W,
    const float* __restrict__ bias, const float* __restrict__ skipGate,
    const float* __restrict__ Hskip, float* __restrict__ Y, int N, int mode)
{
  __shared__ float Wlds[128 * 128]; // 64 KB

  const int tid  = threadIdx.x;
  const int wave = tid >> 5;
  const int lane = tid & 31;
  const int l15  = lane & 15;
  const int hf   = lane >> 4; // which K-half of the 16x4 fragment this lane holds

  // cooperative staging of W into packed LDS layout (coalesced float4 reads)
  for (int i = tid; i < 128 * 32; i += 256) {
    int k  = i >> 5;
    int n4 = (i & 31) << 2;
    float4 wv = *(const float4*)(W + k * 128 + n4);
    int kg = k >> 2, kr = k & 3;
    float* dst = &Wlds[(kg * 128 + n4) * 4 + kr];
    dst[0] = wv.x; dst[4] = wv.y; dst[8] = wv.z; dst[12] = wv.w;
  }
  __syncthreads();

  const int r0 = blockIdx.x * 128 + wave * 16;
  int arow = r0 + l15; if (arow > N - 1) arow = N - 1; // clamp, stores guarded
  const float* xrow = X + (size_t)arow * 128;

  v8f acc[8];
  #pragma unroll
  for (int t = 0; t < 8; ++t) {
    #pragma unroll
    for (int j = 0; j < 8; ++j) acc[t][j] = 0.f;
  }

  for (int kg = 0; kg < 32; ++kg) {
    // A fragment: lane<16 -> X[m, 4kg..4kg+1]; lane>=16 -> X[m, 4kg+2..3]
    v2f a = *(const v2f*)(xrow + kg * 4 + hf * 2);
    #pragma unroll
    for (int t = 0; t < 8; ++t) {
      v2f b = *(const v2f*)(&Wlds[(kg * 128 + t * 16 + l15) * 4 + hf * 2]);
      acc[t] = __builtin_amdgcn_wmma_f32_16x16x4_f32(
          false, a, false, b, (short)0, acc[t], false, false);
    }
  }

  float ss = 0.f;
  if (mode == 1) ss = 1.f / (1.f + __expf(-skipGate[0]));

  #pragma unroll
  for (int t = 0; t < 8; ++t) {
    int col = t * 16 + l15;
    float bc = bias[col];
    #pragma unroll
    for (int r = 0; r < 8; ++r) {
      int grow = r0 + r + hf * 8; // VGPR r: lanes0-15 -> row r, lanes16-31 -> row r+8
      if (grow < N) {
        float val = acc[t][r] + bc;
        if (mode == 1) {
          float hs = Hskip[(size_t)grow * 128 + col];
          val = ss * val + (1.f - ss) * hs;
          val = fmaxf(val, 0.f);
        }
        Y[(size_t)grow * 128 + col] = val;
      }
    }
  }
}

// =====================================================================
// Fold per-head relation matrix into projection weight+bias:
//   Wc[r, h*32+f] = sum_d W[r, h*32+d] * rel[h,d,f]   (r<128)
//   bc[h*32+f]    = sum_d b[h*32+d]    * rel[h,d,f]
// =====================================================================
__global__ __launch_bounds__(256) void combine_rel_kernel(
    const float* __restrict__ W, const float* __restrict__ b,
    const float* __restrict__ rel, float* __restrict__ Wc, float* __restrict__ bc)
{
  int idx = blockIdx.x * 256 + threadIdx.x;
  if (idx >= 129 * 128) return;
  int r = idx >> 7;
  int o = idx & 127;
  int h = o >> 5, f = o & 31;
  const float* src = (r < 128) ? (W + (size_t)r * 128 + h * 32) : (b + h * 32);
  const float* rp  = rel + h * 1024 + f;
  float s = 0.f;
  #pragma unroll 8
  for (int d = 0; d < 32; ++d) s += src[d] * rp[d * 32];
  if (r < 128) Wc[(size_t)r * 128 + o] = s; else bc[o] = s;
}

__global__ void fill_u32_kernel(unsigned* __restrict__ p, unsigned v, int n) {
  int i = blockIdx.x * 256 + threadIdx.x;
  if (i < n) p[i] = v;
}

// =====================================================================
// One wave per edge: alpha[e,h] = scale*p_rel[h]*dot32(KR[src,h,:],Q[dst,h,:])
// and atomicMax of the encoded alpha into per-(dst,head) running max.
// =====================================================================
__global__ __launch_bounds__(256) void alpha_kernel(
    const int* __restrict__ ei, int E,
    const float* __restrict__ KR, const float* __restrict__ Q,
    const float* __restrict__ prel, float scale,
    float* __restrict__ alphaOut, unsigned* __restrict__ mEnc)
{
  int e = blockIdx.x * 8 + (threadIdx.x >> 5);
  if (e >= E) return;
  int lane = threadIdx.x & 31;
  int src = ei[e];
  int dst = ei[E + e];
  float4 k = *(const float4*)(KR + (size_t)src * 128 + lane * 4);
  float4 q = *(const float4*)(Q  + (size_t)dst * 128 + lane * 4);
  float p = k.x * q.x + k.y * q.y + k.z * q.z + k.w * q.w;
  p += __shfl_down(p, 4, 8);
  p += __shfl_down(p, 2, 8);
  p += __shfl_down(p, 1, 8);
  if ((lane & 7) == 0) {
    int h = lane >> 3;
    float a = p * prel[h] * scale;
    alphaOut[(size_t)e * 4 + h] = a;
    atomicMax(mEnc + (size_t)dst * 4 + h, fenc(a));
  }
}

// e_val = exp(alpha - m[dst]); accumulate segment sum with f32 atomics.
__global__ void expsum_kernel(const int* __restrict__ ei, int E,
    float* __restrict__ alpha, const unsigned* __restrict__ mEnc,
    float* __restrict__ ssum)
{
  int i = blockIdx.x * 256 + threadIdx.x;
  if (i >= E * 4) return;
  int e = i >> 2, h = i & 3;
  int dst = ei[E + e];
  float m = fdec(mEnc[(size_t)dst * 4 + h]);
  if (!(m >= -3.4e38f && m <= 3.4e38f)) m = 0.f; // isolated-node guard
  float w = __expf(alpha[i] - m);
  alpha[i] = w;
  atomicAdd(ssum + (size_t)dst * 4 + h, w);
}

// One wave per edge: agg[dst] += (e/(s+1e-16)) * VM[src]   (f32 atomics)
__global__ __launch_bounds__(256) void agg_kernel(
    const int* __restrict__ ei, int E,
    const float* __restrict__ alpha, const float* __restrict__ ssum,
    const float* __restrict__ VM, float* __restrict__ agg)
{
  int e = blockIdx.x * 8 + (threadIdx.x >> 5);
  if (e >= E) return;
  int lane = threadIdx.x & 31;
  int src = ei[e];
  int dst = ei[E + e];
  int h = lane >> 3;
  float w = alpha[(size_t)e * 4 + h] / (ssum[(size_t)dst * 4 + h] + 1e-16f);
  float4 v = *(const float4*)(VM + (size_t)src * 128 + lane * 4);
  float* o = agg + (size_t)dst * 128 + lane * 4;
  atomicAdd(o + 0, w * v.x);
  atomicAdd(o + 1, w * v.y);
  atomicAdd(o + 2, w * v.z);
  atomicAdd(o + 3, w * v.w);
}

// exact gelu, in place
__global__ void gelu_kernel(float* __restrict__ x, int n) {
  int i = blockIdx.x * 256 + threadIdx.x;
  if (i < n) {
    float v = x[i];
    x[i] = 0.5f * v * (1.f + erff(v * 0.70710678118654752f));
  }
}

// =====================================================================
extern "C" void kernel_launch(void* const* d_in, const int* in_sizes, int n_in,
                              void* d_out, int out_size, void* d_ws, size_t ws_size,
                              hipStream_t stream)
{
  (void)n_in; (void)out_size; (void)ws_size;

  const float* x_firm   = (const float*)d_in[0];
  const float* x_person = (const float*)d_in[1];
  const int* ei_ff = (const int*)d_in[2];
  const int* ei_pf = (const int*)d_in[3];
  const int* ei_fp = (const int*)d_in[4];
  const int NF  = in_sizes[0] / HID;
  const int NP  = in_sizes[1] / HID;
  const int Eff = in_sizes[2] / 2;
  const int Epf = in_sizes[3] / 2;
  const int Efp = in_sizes[4] / 2;

  // param leaves: jax tree flatten (sorted dict keys). Per-layer offsets:
  // 0 aW_f 1 aW_p 2 ab_f 3 ab_p | 4 arel_ff 5 arel_fp 6 arel_pf |
  // 7 kW_f 8 kW_p 9 kb_f 10 kb_p | 11 mrel_ff 12 mrel_fp 13 mrel_pf |
  // 14 prel_ff 15 prel_fp 16 prel_pf | 17 qW_f 18 qW_p 19 qb_f 20 qb_p |
  // 21 skip_f 22 skip_p | 23 vW_f 24 vW_p 25 vb_f 26 vb_p
  auto L = [&](int layer, int off) -> const float* {
    return (const float*)d_in[5 + layer * 27 + off];
  };
  const float* proj_W = (const float*)d_in[5 + 2 * 27 + 0];
  const float* proj_b = (const float*)d_in[5 + 2 * 27 + 1];

  // ---- carve workspace ----
  char* wp = (char*)d_ws;
  auto alloc = [&](size_t bytes) -> void* {
    void* p = (void*)wp;
    wp += (bytes + 255) & ~(size_t)255;
    return p;
  };
  float* hA_f = (float*)alloc((size_t)NF * HID * 4);
  float* hA_p = (float*)alloc((size_t)NP * HID * 4);
  float* hB_f = (float*)alloc((size_t)NF * HID * 4);
  float* hB_p = (float*)alloc((size_t)NP * HID * 4);
  float* Qf   = (float*)alloc((size_t)NF * HID * 4);
  float* Qp   = (float*)alloc((size_t)NP * HID * 4);
  float* T0   = (float*)alloc((size_t)NF * HID * 4); // ff: src firm
  float* T1   = (float*)alloc((size_t)NP * HID * 4); // pf: src person
  float* T2   = (float*)alloc((size_t)NF * HID * 4); // fp: src firm
  float* aggF = (float*)alloc((size_t)NF * HID * 4);
  float* aggP = (float*)alloc((size_t)NP * HID * 4);
  float* alphaF = (float*)alloc((size_t)(Eff + Epf) * 4 * 4);
  float* alphaP = (float*)alloc((size_t)Efp * 4 * 4);
  unsigned* mF = (unsigned*)alloc((size_t)NF * 4 * 4);
  unsigned* mP = (unsigned*)alloc((size_t)NP * 4 * 4);
  float* sF = (float*)alloc((size_t)NF * 4 * 4);
  float* sP = (float*)alloc((size_t)NP * 4 * 4);
  float* WcKA[3]; float* WcVM[3]; float* bcKA[3]; float* bcVM[3];
  for (int i = 0; i < 3; ++i) {
    WcKA[i] = (float*)alloc(128 * 128 * 4);
    WcVM[i] = (float*)alloc(128 * 128 * 4);
    bcKA[i] = (float*)alloc(128 * 4);
    bcVM[i] = (float*)alloc(128 * 4);
  }

  const float scale = 0.17677669529663687f; // 1/sqrt(32)
  const int gF = (NF + 127) / 128;
  const int gP = (NP + 127) / 128;
  const int gC = (129 * 128 + 255) / 256;

  const float* hin_f = x_firm;
  const float* hin_p = x_person;
  float* hout_f = hA_f;
  float* hout_p = hA_p;

  for (int layer = 0; layer < 2; ++layer) {
    // 1) fold relation matrices into projection weights
    combine_rel_kernel<<<gC, 256, 0, stream>>>(L(layer,7),  L(layer,9),  L(layer,4),  WcKA[0], bcKA[0]); // ff key
    combine_rel_kernel<<<gC, 256, 0, stream>>>(L(layer,8),  L(layer,10), L(layer,6),  WcKA[1], bcKA[1]); // pf key
    combine_rel_kernel<<<gC, 256, 0, stream>>>(L(layer,7),  L(layer,9),  L(layer,5),  WcKA[2], bcKA[2]); // fp key
    combine_rel_kernel<<<gC, 256, 0, stream>>>(L(layer,23), L(layer,25), L(layer,11), WcVM[0], bcVM[0]); // ff val
    combine_rel_kernel<<<gC, 256, 0, stream>>>(L(layer,24), L(layer,26), L(layer,13), WcVM[1], bcVM[1]); // pf val
    combine_rel_kernel<<<gC, 256, 0, stream>>>(L(layer,23), L(layer,25), L(layer,12), WcVM[2], bcVM[2]); // fp val

    // 2) queries
    gemm128_kernel<<<gF, 256, 0, stream>>>(hin_f, L(layer,17), L(layer,19), nullptr, nullptr, Qf, NF, 0);
    gemm128_kernel<<<gP, 256, 0, stream>>>(hin_p, L(layer,18), L(layer,20), nullptr, nullptr, Qp, NP, 0);

    // 3) relation-transformed keys (node-level)
    gemm128_kernel<<<gF, 256, 0, stream>>>(hin_f, WcKA[0], bcKA[0], nullptr, nullptr, T0, NF, 0);
    gemm128_kernel<<<gP, 256, 0, stream>>>(hin_p, WcKA[1], bcKA[1], nullptr, nullptr, T1, NP, 0);
    gemm128_kernel<<<gF, 256, 0, stream>>>(hin_f, WcKA[2], bcKA[2], nullptr, nullptr, T2, NF, 0);

    // 4) init segment-max / segment-sum / aggregation buffers
    fill_u32_kernel<<<(NF*4   + 255) / 256, 256, 0, stream>>>(mF, ENC_NEG_INF, NF * 4);
    fill_u32_kernel<<<(NP*4   + 255) / 256, 256, 0, stream>>>(mP, ENC_NEG_INF, NP * 4);
    fill_u32_kernel<<<(NF*4   + 255) / 256, 256, 0, stream>>>((unsigned*)sF, 0u, NF * 4);
    fill_u32_kernel<<<(NP*4   + 255) / 256, 256, 0, stream>>>((unsigned*)sP, 0u, NP * 4);
    fill_u32_kernel<<<(NF*128 + 255) / 256, 256, 0, stream>>>((unsigned*)aggF, 0u, NF * 128);
    fill_u32_kernel<<<(NP*128 + 255) / 256, 256, 0, stream>>>((unsigned*)aggP, 0u, NP * 128);

    // 5) attention logits + running max
    alpha_kernel<<<(Eff + 7) / 8, 256, 0, stream>>>(ei_ff, Eff, T0, Qf, L(layer,14), scale, alphaF, mF);
    alpha_kernel<<<(Epf + 7) / 8, 256, 0, stream>>>(ei_pf, Epf, T1, Qf, L(layer,16), scale, alphaF + (size_t)Eff * 4, mF);
    alpha_kernel<<<(Efp + 7) / 8, 256, 0, stream>>>(ei_fp, Efp, T2, Qp, L(layer,15), scale, alphaP, mP);

    // 6) relation-transformed values (reuse T buffers; keys no longer needed)
    gemm128_kernel<<<gF, 256, 0, stream>>>(hin_f, WcVM[0], bcVM[0], nullptr, nullptr, T0, NF, 0);
    gemm128_kernel<<<gP, 256, 0, stream>>>(hin_p, WcVM[1], bcVM[1], nullptr, nullptr, T1, NP, 0);
    gemm128_kernel<<<gF, 256, 0, stream>>>(hin_f, WcVM[2], bcVM[2], nullptr, nullptr, T2, NF, 0);

    // 7) exp + segment sum
    expsum_kernel<<<(Eff*4 + 255) / 256, 256, 0, stream>>>(ei_ff, Eff, alphaF, mF, sF);
    expsum_kernel<<<(Epf*4 + 255) / 256, 256, 0, stream>>>(ei_pf, Epf, alphaF + (size_t)Eff * 4, mF, sF);
    expsum_kernel<<<(Efp*4 + 255) / 256, 256, 0, stream>>>(ei_fp, Efp, alphaP, mP, sP);

    // 8) weighted message aggregation
    agg_kernel<<<(Eff + 7) / 8, 256, 0, stream>>>(ei_ff, Eff, alphaF, sF, T0, aggF);
    agg_kernel<<<(Epf + 7) / 8, 256, 0, stream>>>(ei_pf, Epf, alphaF + (size_t)Eff * 4, sF, T1, aggF);
    agg_kernel<<<(Efp + 7) / 8, 256, 0, stream>>>(ei_fp, Efp, alphaP, sP, T2, aggP);

    // 9) gelu(agg)
    gelu_kernel<<<(NF*128 + 255) / 256, 256, 0, stream>>>(aggF, NF * 128);
    gelu_kernel<<<(NP*128 + 255) / 256, 256, 0, stream>>>(aggP, NP * 128);

    // 10) output projection + gated skip + relu
    gemm128_kernel<<<gF, 256, 0, stream>>>(aggF, L(layer,0), L(layer,2), L(layer,21), hin_f, hout_f, NF, 1);
    gemm128_kernel<<<gP, 256, 0, stream>>>(aggP, L(layer,1), L(layer,3), L(layer,22), hin_p, hout_p, NP, 1);

    hin_f = hout_f; hin_p = hout_p;
    hout_f = hB_f;  hout_p = hB_p;
  }

  // final projection on firm features
  gemm128_kernel<<<gF, 256, 0, stream>>>(hin_f, proj_W, proj_b, nullptr, nullptr, (float*)d_out, NF, 0);
}